// RegressLoss_17033840295996
// MI455X (gfx1250) — compile-verified
//
#include <hip/hip_runtime.h>

#define DEGF 0.017453292519943295f
#define IOU_THRESH 0.5f
#define IND_THRESH 0.1f
#define BETA (1.0f/9.0f)

typedef __attribute__((ext_vector_type(2))) float v2f;
typedef __attribute__((ext_vector_type(8))) float v8f;

__device__ __forceinline__ float crossf(float ax, float ay, float bx, float by) {
    return ax * by - ay * bx;
}

__device__ __forceinline__ void boxCorners(float x1, float y1, float x2, float y2,
                                           float tdeg, float* cxo, float* cyo) {
    float cx = 0.5f * (x1 + x2), cy = 0.5f * (y1 + y2);
    float hw = 0.5f * (x2 - x1), hh = 0.5f * (y2 - y1);
    float th = tdeg * DEGF;
    float c = cosf(th), s = sinf(th);
    const float dx[4] = {-1.f, 1.f, 1.f, -1.f};
    const float dy[4] = {-1.f, -1.f, 1.f, 1.f};
#pragma unroll
    for (int k = 0; k < 4; ++k) {
        float ddx = dx[k] * hw, ddy = dy[k] * hh;
        cxo[k] = cx + c * ddx - s * ddy;
        cyo[k] = cy + s * ddx + c * ddy;
    }
}

// Convex intersection area of two rotated quads, matching the reference
// _pair_inter_area (24 candidate points, stable angle sort, shoelace).
__device__ float pairInterArea(const float* px, const float* py,
                               const float* qx, const float* qy) {
    float ptsx[24], ptsy[24];
    bool val[24];

    // P corners inside Q
#pragma unroll
    for (int i = 0; i < 4; ++i) {
        ptsx[i] = px[i]; ptsy[i] = py[i];
        bool all_in = true;
#pragma unroll
        for (int e = 0; e < 4; ++e) {
            int e1 = (e + 1) & 3;
            float cr = crossf(qx[e1] - qx[e], qy[e1] - qy[e],
                              px[i] - qx[e], py[i] - qy[e]);
            all_in = all_in && (cr >= -1e-6f);
        }
        val[i] = all_in;
    }
    // Q corners inside P
#pragma unroll
    for (int i = 0; i < 4; ++i) {
        ptsx[4 + i] = qx[i]; ptsy[4 + i] = qy[i];
        bool all_in = true;
#pragma unroll
        for (int e = 0; e < 4; ++e) {
            int e1 = (e + 1) & 3;
            float cr = crossf(px[e1] - px[e], py[e1] - py[e],
                              qx[i] - px[e], qy[i] - py[e]);
            all_in = all_in && (cr >= -1e-6f);
        }
        val[4 + i] = all_in;
    }
    // Edge-edge intersections (i over P edges outer, j over Q edges inner
    // to match xpts.reshape(16) ordering)
    int k = 8;
    for (int i = 0; i < 4; ++i) {
        int i1 = (i + 1) & 3;
        float rx = px[i1] - px[i], ry = py[i1] - py[i];
        for (int j = 0; j < 4; ++j) {
            int j1 = (j + 1) & 3;
            float sx = qx[j1] - qx[j], sy = qy[j1] - qy[j];
            float qpx = qx[j] - px[i], qpy = qy[j] - py[i];
            float den = crossf(rx, ry, sx, sy);
            bool dok = fabsf(den) >= 1e-8f;
            float safe = dok ? den : 1.0f;
            float t = crossf(qpx, qpy, sx, sy) / safe;
            float u = crossf(qpx, qpy, rx, ry) / safe;
            bool ok = dok && (t >= 0.f) && (t <= 1.f) && (u >= 0.f) && (u <= 1.f);
            ptsx[k] = px[i] + t * rx;
            ptsy[k] = py[i] + t * ry;
            val[k] = ok;
            ++k;
        }
    }

    int n = 0;
    float cx = 0.f, cy = 0.f;
    for (int m = 0; m < 24; ++m) {
        if (val[m]) { ++n; cx += ptsx[m]; cy += ptsy[m]; }
    }
    int nmax = (n > 1) ? n : 1;
    cx = cx / (float)nmax;
    cy = cy / (float)nmax;

    float ang[24], rx2[24], ry2[24];
    for (int m = 0; m < 24; ++m) {
        float rxm = ptsx[m] - cx, rym = ptsy[m] - cy;
        rx2[m] = rxm; ry2[m] = rym;
        ang[m] = val[m] ? atan2f(rym, rxm) : 1e9f;
    }
    // Stable insertion sort by angle (mirrors jnp stable argsort)
    for (int m = 1; m < 24; ++m) {
        float ka = ang[m], kx = rx2[m], ky = ry2[m];
        int j2 = m - 1;
        while (j2 >= 0 && ang[j2] > ka) {
            ang[j2 + 1] = ang[j2]; rx2[j2 + 1] = rx2[j2]; ry2[j2 + 1] = ry2[j2];
            --j2;
        }
        ang[j2 + 1] = ka; rx2[j2 + 1] = kx; ry2[j2 + 1] = ky;
    }
    float area2 = 0.f;
    for (int idx = 0; idx < 24; ++idx) {
        if (idx < n) {
            int nx = (idx + 1 < n) ? idx + 1 : 0;
            area2 += crossf(rx2[idx], ry2[idx], rx2[nx], ry2[nx]);
        }
    }
    float area = 0.5f * fabsf(area2);
    return (n >= 3) ? area : 0.0f;
}

__global__ void rloss_init(float* partials, int count) {
    int i = blockIdx.x * blockDim.x + threadIdx.x;
    if (i < count) partials[i] = 0.0f;
}

// One thread per anchor; 64 blocks of 256 threads per image.
// partials layout: [0 .. B*64)       loss partial per (img, block)
//                  [B*64 .. 2*B*64)  pos-count partial per (img, block)
__global__ void rloss_main(const float* __restrict__ reg,
                           const float* __restrict__ anc,
                           const float* __restrict__ ann,
                           float* __restrict__ partials,
                           int N, int B) {
    const int M = 16;
    int img = blockIdx.x >> 6;
    int blk = blockIdx.x & 63;
    int tid = threadIdx.x;

    __shared__ float sAnn[M * 6];
    __shared__ float sQx[M][4];
    __shared__ float sQy[M][4];
    __shared__ float sSq[M][4];
    __shared__ float sAreaG[M];

    if (tid < M * 6) sAnn[tid] = ann[img * M * 6 + tid];
    __syncthreads();
    if (tid < M) {
        float x1 = sAnn[tid * 6 + 0], y1 = sAnn[tid * 6 + 1];
        float x2 = sAnn[tid * 6 + 2], y2 = sAnn[tid * 6 + 3];
        float t  = sAnn[tid * 6 + 4];
        float qcx[4], qcy[4];
        boxCorners(x1, y1, x2, y2, t, qcx, qcy);
#pragma unroll
        for (int k = 0; k < 4; ++k) { sQx[tid][k] = qcx[k]; sQy[tid][k] = qcy[k]; }
        float w = x2 - x1, h = y2 - y1;
        float ccx = x1 + 0.5f * w, ccy = y1 + 0.5f * h;
        float s = 0.5f * fmaxf(w, h);
        sSq[tid][0] = ccx - s; sSq[tid][1] = ccy - s;
        sSq[tid][2] = ccx + s; sSq[tid][3] = ccy + s;
        sAreaG[tid] = w * h;
    }
    __syncthreads();

    float lsum = 0.0f;
    float pcnt = 0.0f;

    for (int i = blk * blockDim.x + tid; i < N; i += 64 * blockDim.x) {
        const float* a = anc + (size_t)(img * N + i) * 5;
        float ax1 = a[0], ay1 = a[1], ax2 = a[2], ay2 = a[3], at = a[4];
        float pxx[4], pyy[4];
        boxCorners(ax1, ay1, ax2, ay2, at, pxx, pyy);
        float aw = ax2 - ax1, ah = ay2 - ay1;
        float acx = ax1 + 0.5f * aw, acy = ay1 + 0.5f * ah;
        float as = 0.5f * fmaxf(aw, ah);
        float sqx1 = acx - as, sqy1 = acy - as, sqx2 = acx + as, sqy2 = acy + as;
        float areaA = aw * ah;
        float areaSqA = (sqx2 - sqx1) * (sqy2 - sqy1);

        float best = -3.402823466e38f;
        int arg = 0;
        for (int j = 0; j < M; ++j) {
            // axis-aligned indicator IoU on min-area squares
            float lx = fmaxf(sqx1, sSq[j][0]);
            float ly = fmaxf(sqy1, sSq[j][1]);
            float rx = fminf(sqx2, sSq[j][2]);
            float ry = fminf(sqy2, sSq[j][3]);
            float iw = fmaxf(rx - lx, 0.f), ih = fmaxf(ry - ly, 0.f);
            float sInter = iw * ih;
            float areaSqG = (sSq[j][2] - sSq[j][0]) * (sSq[j][3] - sSq[j][1]);
            float ind = sInter / fmaxf(areaSqA + areaSqG - sInter, 1e-8f);

            float iou = 0.0f;
            if (ind > IND_THRESH) {
                float qcx[4], qcy[4];
#pragma unroll
                for (int k = 0; k < 4; ++k) { qcx[k] = sQx[j][k]; qcy[k] = sQy[j][k]; }
                float inter = pairInterArea(pxx, pyy, qcx, qcy);
                iou = inter / fmaxf(areaA + sAreaG[j] - inter, 1e-8f);
            }
            bool validj = sAnn[j * 6 + 5] != -1.0f;
            float ov = validj ? iou : -1.0f;
            if (ov > best) { best = ov; arg = j; }   // strict > => first argmax
        }

        if (best >= IOU_THRESH) {
            const float* g = sAnn + arg * 6;
            float ew = fmaxf(ax2 - ax1, 1.0f), eh = fmaxf(ay2 - ay1, 1.0f);
            float ecx = ax1 + 0.5f * ew, ecy = ay1 + 0.5f * eh;
            float gw = fmaxf(g[2] - g[0], 1.0f), gh = fmaxf(g[3] - g[1], 1.0f);
            float gcx = g[0] + 0.5f * gw, gcy = g[1] + 0.5f * gh;
            float tgt[5];
            tgt[0] = 10.0f * (gcx - ecx) / ew;
            tgt[1] = 10.0f * (gcy - ecy) / eh;
            tgt[2] = 5.0f * logf(gw / ew);
            tgt[3] = 5.0f * logf(gh / eh);
            tgt[4] = 15.0f * (tanf(g[4] * DEGF) - tanf(at * DEGF));
            const float* r = reg + (size_t)(img * N + i) * 5;
            float s5 = 0.0f;
#pragma unroll
            for (int kk = 0; kk < 5; ++kk) {
                float d = fabsf(r[kk] - tgt[kk]);
                s5 += (d < BETA) ? (0.5f * d * d / BETA) : (d - 0.5f * BETA);
            }
            lsum += s5;
            pcnt += 1.0f;
        }
    }

    __shared__ float rl[256];
    __shared__ float rc[256];
    rl[tid] = lsum; rc[tid] = pcnt;
    __syncthreads();
    for (int s2 = 128; s2 > 0; s2 >>= 1) {
        if (tid < s2) { rl[tid] += rl[tid + s2]; rc[tid] += rc[tid + s2]; }
        __syncthreads();
    }
    if (tid == 0) {
        partials[img * 64 + blk]        = rl[0];
        partials[(B + img) * 64 + blk]  = rc[0];
    }
}

// Sum 64 fp32 values with one V_WMMA_F32_16X16X4_F32:
// A = the 64 partials packed 2/lane (16x4 fp32 A-matrix), B = all-ones.
// D rows hold per-M sums replicated across columns; per-lane sum of the 8 D
// VGPRs covers rows 0-7 (lanes 0-15) / 8-15 (lanes 16-31); xor-16 shuffle
// combines them into the grand total in every lane. EXEC is all ones here.
__device__ __forceinline__ float wmma_sum64(const float* p) {
    int lane = threadIdx.x & 31;
    v2f a; a.x = p[2 * lane]; a.y = p[2 * lane + 1];
    v2f b; b.x = 1.0f; b.y = 1.0f;
    v8f c = {0.f, 0.f, 0.f, 0.f, 0.f, 0.f, 0.f, 0.f};
    v8f d = __builtin_amdgcn_wmma_f32_16x16x4_f32(
        false, a, false, b, (short)0, c, false, false);
    float s = d[0] + d[1] + d[2] + d[3] + d[4] + d[5] + d[6] + d[7];
    s += __shfl_xor(s, 16, 32);
    return s;
}

__global__ void rloss_finalize(const float* __restrict__ partials,
                               float* __restrict__ out, int B) {
    float acc = 0.0f;
    for (int b = 0; b < B; ++b) {
        float lossSum = wmma_sum64(partials + b * 64);
        float cntSum  = wmma_sum64(partials + (B + b) * 64);
        float l = (cntSum >= 1.0f) ? (lossSum / (cntSum * 5.0f)) : 0.0f;
        acc += l;
    }
    if (threadIdx.x == 0) out[0] = acc / (float)B;
}

extern "C" void kernel_launch(void* const* d_in, const int* in_sizes, int n_in,
                              void* d_out, int out_size, void* d_ws, size_t ws_size,
                              hipStream_t stream) {
    const float* reg = (const float*)d_in[0];   // (B, N, 5)
    const float* anc = (const float*)d_in[1];   // (B, N, 5)
    const float* ann = (const float*)d_in[2];   // (B, 16, 6)
    float* out = (float*)d_out;                 // (1,)
    float* partials = (float*)d_ws;             // 2*B*64 floats

    int B = in_sizes[2] / (16 * 6);
    if (B < 1) B = 1;
    int N = in_sizes[1] / (B * 5);

    int count = 2 * B * 64;
    rloss_init<<<(count + 255) / 256, 256, 0, stream>>>(partials, count);
    rloss_main<<<dim3(B * 64), dim3(256), 0, stream>>>(reg, anc, ann, partials, N, B);
    rloss_finalize<<<1, 32, 0, stream>>>(partials, out, B);
}